// DGCNN_9113920602560
// MI455X (gfx1250) — compile-verified
//
#include <hip/hip_runtime.h>

typedef float v2f __attribute__((ext_vector_type(2)));
typedef float v8f __attribute__((ext_vector_type(8)));

#define BATCH 8
#define NPTS  1024
#define KNN_K 20
#define NKTOT (NPTS * KNN_K)   // 20480; /16 = 1280 tiles; /32 = 640 tile-pairs

// ---------------------------------------------------------------------------
// Prep: transpose x0 (B,3,N) -> x0t (B,N,4) zero-padded; pad w1 (64,6) ->
// wp (64,8) with zeros at channels 3 and 7 so layer 1 runs the generic C=4 path.
// ---------------------------------------------------------------------------
__global__ void prep_kernel(const float* __restrict__ x0, const float* __restrict__ w1,
                            float* __restrict__ x0t, float* __restrict__ wp)
{
    int t = blockIdx.x * blockDim.x + threadIdx.x;
    if (t < BATCH * NPTS) {
        int b = t >> 10, n = t & 1023;
        x0t[t * 4 + 0] = x0[(b * 3 + 0) * NPTS + n];
        x0t[t * 4 + 1] = x0[(b * 3 + 1) * NPTS + n];
        x0t[t * 4 + 2] = x0[(b * 3 + 2) * NPTS + n];
        x0t[t * 4 + 3] = 0.f;
    }
    if (t < 64) {
        wp[t * 8 + 0] = w1[t * 6 + 0];
        wp[t * 8 + 1] = w1[t * 6 + 1];
        wp[t * 8 + 2] = w1[t * 6 + 2];
        wp[t * 8 + 3] = 0.f;
        wp[t * 8 + 4] = w1[t * 6 + 3];
        wp[t * 8 + 5] = w1[t * 6 + 4];
        wp[t * 8 + 6] = w1[t * 6 + 5];
        wp[t * 8 + 7] = 0.f;
    }
}

// ---------------------------------------------------------------------------
// KNN on point-major features x (B,N,C): top-20 of
// pd[i][j] = 2*x_i.x_j - |x_i|^2 - |x_j|^2.  Block = 128 query points,
// j staged through LDS in tiles of 64 points.  C is a power of two.
// ---------------------------------------------------------------------------
template <int C>
__global__ void knn_kernel(const float* __restrict__ x, int* __restrict__ idx)
{
    __shared__ float xs[64 * C];
    __shared__ float x2s[64];

    const int t     = threadIdx.x;       // 0..127
    const int b     = blockIdx.x >> 3;
    const int itile = blockIdx.x & 7;
    const int i     = itile * 128 + t;

    const float* xb = x + (size_t)b * NPTS * C;

    float xi[C];
#pragma unroll
    for (int c = 0; c < C; ++c) xi[c] = xb[(size_t)i * C + c];
    float x2i = 0.f;
#pragma unroll
    for (int c = 0; c < C; ++c) x2i += xi[c] * xi[c];

    float kd[KNN_K];
    int   ki[KNN_K];
#pragma unroll
    for (int s = 0; s < KNN_K; ++s) { kd[s] = -3.4e38f; ki[s] = 0; }

    for (int jt = 0; jt < NPTS / 64; ++jt) {
        __syncthreads();
        for (int u = t; u < 64 * C; u += 128)
            xs[u] = xb[(size_t)jt * 64 * C + u];   // point-major tile copy
        __syncthreads();
        if (t < 64) {
            float s = 0.f;
#pragma unroll
            for (int c = 0; c < C; ++c) { float v = xs[t * C + c]; s += v * v; }
            x2s[t] = s;
        }
        __syncthreads();
        for (int jj = 0; jj < 64; ++jj) {
            float dot = 0.f;
#pragma unroll
            for (int c = 0; c < C; ++c) dot += xi[c] * xs[jj * C + c];
            float d = 2.f * dot - x2i - x2s[jj];
            int   j = jt * 64 + jj;
#pragma unroll
            for (int s = 0; s < KNN_K; ++s) {       // stable descending insert
                if (d > kd[s]) {
                    float td = kd[s]; int ti = ki[s];
                    kd[s] = d; ki[s] = j;
                    d = td; j = ti;
                }
            }
        }
    }
#pragma unroll
    for (int s = 0; s < KNN_K; ++s) idx[(b * NPTS + i) * KNN_K + s] = ki[s];
}

// ---------------------------------------------------------------------------
// EdgeConv GEMM (branch-free): D(16o x 16nk) = W(16o x 2C) * F(2C x 16nk)
// with F = [center ; nbr - center], fp32 WMMA 16x16x4, 2 nk-tiles per wave.
// apply==0: per-channel sum/sumsq (BN stats).  apply==1: BN+ReLU+max-over-K.
// ---------------------------------------------------------------------------
template <int C>
__global__ void edgeconv_gemm(const float* __restrict__ x,   // (B,N,C)
                              const int*   __restrict__ nbr, // (B,N,K)
                              const float* __restrict__ W,   // (O,2C)
                              const float* __restrict__ p0,  // scale (apply)
                              const float* __restrict__ p1,  // shift (apply)
                              float* __restrict__ a0,        // sum   (stats)
                              float* __restrict__ a1,        // sumsq (stats)
                              unsigned int* __restrict__ out,// (B,N,O)
                              int O, int apply)
{
    constexpr int C2 = 2 * C;
    const int wid  = (blockIdx.x * blockDim.x + threadIdx.x) >> 5;
    const int lane = threadIdx.x & 31;
    const int half = lane >> 4;
    const int col  = lane & 15;

    const int otiles = O >> 4;
    const int otile  = wid % otiles;
    int       rest   = wid / otiles;
    const int pair   = rest % (NKTOT / 32);
    const int b      = rest / (NKTOT / 32);

    const float* xb = x + (size_t)b * NPTS * C;
    const float* Wr = W + (size_t)(otile * 16 + col) * C2;   // A: M = lane%16

    int nn[2], jn[2];
#pragma unroll
    for (int t2 = 0; t2 < 2; ++t2) {
        int nk = (pair * 2 + t2) * 16 + col;
        nn[t2] = nk / KNN_K;
        jn[t2] = nbr[(b * NPTS + nn[t2]) * KNN_K + (nk % KNN_K)];
    }
    const float* pn0 = xb + (size_t)nn[0] * C;
    const float* pn1 = xb + (size_t)nn[1] * C;
    const float* pj0 = xb + (size_t)jn[0] * C;
    const float* pj1 = xb + (size_t)jn[1] * C;

    v8f acc0 = {}, acc1 = {};

    // ---- center phase: F[c] = x[n][c], c in [0,C) ----
#pragma unroll 4
    for (int cs = 0; cs < C / 4; ++cs) {
        const int c0 = cs * 4 + 2 * half;   // lanes 0-15: K={0,1}; 16-31: K={2,3}
        v2f av = *(const v2f*)(Wr + c0);
        v2f b0 = *(const v2f*)(pn0 + c0);
        v2f b1 = *(const v2f*)(pn1 + c0);
        acc0 = __builtin_amdgcn_wmma_f32_16x16x4_f32(false, av, false, b0, (short)0, acc0, false, false);
        acc1 = __builtin_amdgcn_wmma_f32_16x16x4_f32(false, av, false, b1, (short)0, acc1, false, false);
    }
    // ---- diff phase: F[C+c] = x[j][c] - x[n][c] ----
#pragma unroll 4
    for (int cs = 0; cs < C / 4; ++cs) {
        const int c0 = cs * 4 + 2 * half;
        v2f av  = *(const v2f*)(Wr + C + c0);
        v2f n0v = *(const v2f*)(pn0 + c0);
        v2f n1v = *(const v2f*)(pn1 + c0);
        v2f j0v = *(const v2f*)(pj0 + c0);
        v2f j1v = *(const v2f*)(pj1 + c0);
        v2f b0; b0.x = j0v.x - n0v.x; b0.y = j0v.y - n0v.y;
        v2f b1; b1.x = j1v.x - n1v.x; b1.y = j1v.y - n1v.y;
        acc0 = __builtin_amdgcn_wmma_f32_16x16x4_f32(false, av, false, b0, (short)0, acc0, false, false);
        acc1 = __builtin_amdgcn_wmma_f32_16x16x4_f32(false, av, false, b1, (short)0, acc1, false, false);
    }

    if (!apply) {
#pragma unroll
        for (int r = 0; r < 8; ++r) {
            float v  = acc0[r] + acc1[r];
            float v2 = acc0[r] * acc0[r] + acc1[r] * acc1[r];
#pragma unroll
            for (int off = 1; off < 16; off <<= 1) {
                v  += __shfl_xor(v,  off, 32);
                v2 += __shfl_xor(v2, off, 32);
            }
            if (col == 0) {
                int o = otile * 16 + r + 8 * half;   // D: M = r + 8*(lane/16)
                atomicAdd(&a0[o], v);
                atomicAdd(&a1[o], v2);
            }
        }
    } else {
#pragma unroll
        for (int r = 0; r < 8; ++r) {
            int   o  = otile * 16 + r + 8 * half;
            float s  = p0[o], sh = p1[o];
            float v0 = fmaxf(fmaf(acc0[r], s, sh), 0.f);
            float v1 = fmaxf(fmaf(acc1[r], s, sh), 0.f);
            atomicMax(&out[(size_t)(b * NPTS + nn[0]) * O + o], __float_as_uint(v0));
            atomicMax(&out[(size_t)(b * NPTS + nn[1]) * O + o], __float_as_uint(v1));
        }
    }
}

// ---------------------------------------------------------------------------
// Final GEMM: h5 = W5(1024x512) * cat(x1..x4)(512 x B*N); four uniform
// sub-loops (one per source buffer), 2 column-tiles per wave.
// ---------------------------------------------------------------------------
template <int CP>
__device__ __forceinline__ void final_phase(const float* __restrict__ xp, int off,
                                            const float* __restrict__ Wr,
                                            int bn0, int bn1, int half,
                                            v8f& acc0, v8f& acc1)
{
    const float* pa = xp + (size_t)bn0 * CP;
    const float* pb = xp + (size_t)bn1 * CP;
#pragma unroll 4
    for (int cs = 0; cs < CP / 4; ++cs) {
        const int c0 = cs * 4 + 2 * half;
        v2f av = *(const v2f*)(Wr + off + c0);
        v2f b0 = *(const v2f*)(pa + c0);
        v2f b1 = *(const v2f*)(pb + c0);
        acc0 = __builtin_amdgcn_wmma_f32_16x16x4_f32(false, av, false, b0, (short)0, acc0, false, false);
        acc1 = __builtin_amdgcn_wmma_f32_16x16x4_f32(false, av, false, b1, (short)0, acc1, false, false);
    }
}

__global__ void final_gemm(const float* __restrict__ x1, const float* __restrict__ x2,
                           const float* __restrict__ x3, const float* __restrict__ x4,
                           const float* __restrict__ W,
                           const float* __restrict__ p0, const float* __restrict__ p1,
                           float* __restrict__ a0, float* __restrict__ a1,
                           unsigned int* __restrict__ out, int apply)
{
    const int wid  = (blockIdx.x * blockDim.x + threadIdx.x) >> 5;
    const int lane = threadIdx.x & 31;
    const int half = lane >> 4;
    const int col  = lane & 15;

    const int otile = wid % 64;          // 1024/16
    const int pair  = wid / 64;          // 0..255 over (B*N/32)

    const int nc0 = (pair * 2) * 16 + col;
    const int nc1 = nc0 + 16;            // same b: 1024 % 32 == 0
    const int b   = nc0 >> 10;
    const int bn0 = b * NPTS + (nc0 & 1023);
    const int bn1 = b * NPTS + (nc1 & 1023);

    const float* Wr = W + (size_t)(otile * 16 + col) * 512;

    v8f acc0 = {}, acc1 = {};
    final_phase<64> (x1,   0, Wr, bn0, bn1, half, acc0, acc1);
    final_phase<64> (x2,  64, Wr, bn0, bn1, half, acc0, acc1);
    final_phase<128>(x3, 128, Wr, bn0, bn1, half, acc0, acc1);
    final_phase<256>(x4, 256, Wr, bn0, bn1, half, acc0, acc1);

    if (!apply) {
#pragma unroll
        for (int r = 0; r < 8; ++r) {
            float v  = acc0[r] + acc1[r];
            float v2 = acc0[r] * acc0[r] + acc1[r] * acc1[r];
#pragma unroll
            for (int off = 1; off < 16; off <<= 1) {
                v  += __shfl_xor(v,  off, 32);
                v2 += __shfl_xor(v2, off, 32);
            }
            if (col == 0) {
                int o = otile * 16 + r + 8 * half;
                atomicAdd(&a0[o], v);
                atomicAdd(&a1[o], v2);
            }
        }
    } else {
#pragma unroll
        for (int r = 0; r < 8; ++r) {
            int   o  = otile * 16 + r + 8 * half;
            float s  = p0[o], sh = p1[o];
            float v0 = fmaxf(fmaf(acc0[r], s, sh), 0.f);
            float v1 = fmaxf(fmaf(acc1[r], s, sh), 0.f);
            atomicMax(&out[b * 1024 + o], __float_as_uint(v0));
            atomicMax(&out[b * 1024 + o], __float_as_uint(v1));
        }
    }
}

// ---------------------------------------------------------------------------
__global__ void bn_finalize(const float* __restrict__ sum, const float* __restrict__ sumsq,
                            const float* __restrict__ g, const float* __restrict__ bb,
                            float* __restrict__ scale, float* __restrict__ shift,
                            int O, float invcnt)
{
    int o = blockIdx.x * blockDim.x + threadIdx.x;
    if (o >= O) return;
    float m   = sum[o] * invcnt;
    float var = fmaxf(sumsq[o] * invcnt - m * m, 0.f);
    float s   = g[o] * rsqrtf(var + 1e-5f);
    scale[o] = s;
    shift[o] = bb[o] - m * s;
}

// ---------------------------------------------------------------------------
extern "C" void kernel_launch(void* const* d_in, const int* in_sizes, int n_in,
                              void* d_out, int out_size, void* d_ws, size_t ws_size,
                              hipStream_t stream)
{
    const float* x0 = (const float*)d_in[0];
    const float* w1 = (const float*)d_in[1];
    const float* g1 = (const float*)d_in[2];
    const float* b1 = (const float*)d_in[3];
    const float* w2 = (const float*)d_in[4];
    const float* g2 = (const float*)d_in[5];
    const float* b2 = (const float*)d_in[6];
    const float* w3 = (const float*)d_in[7];
    const float* g3 = (const float*)d_in[8];
    const float* b3 = (const float*)d_in[9];
    const float* w4 = (const float*)d_in[10];
    const float* g4 = (const float*)d_in[11];
    const float* b4 = (const float*)d_in[12];
    const float* w5 = (const float*)d_in[13];
    const float* g5 = (const float*)d_in[14];
    const float* b5 = (const float*)d_in[15];

    float* wsf = (float*)d_ws;
    float* x0t = wsf;                                   // 8*1024*4
    float* wp  = x0t + (size_t)BATCH * NPTS * 4;        // 64*8
    float* x1  = wp  + 64 * 8;                          // (B,N,64)
    float* x2  = x1  + (size_t)BATCH * NPTS * 64;
    float* x3  = x2  + (size_t)BATCH * NPTS * 64;       // (B,N,128)
    float* x4  = x3  + (size_t)BATCH * NPTS * 128;      // (B,N,256)
    int*   nb  = (int*)(x4 + (size_t)BATCH * NPTS * 256);
    float* s0  = (float*)(nb + (size_t)BATCH * NPTS * KNN_K);
    float* s1  = s0 + 1024;
    float* sc  = s1 + 1024;
    float* sh  = sc + 1024;

    const float inv_nk = 1.f / (float)((size_t)BATCH * NKTOT);
    const float inv_n  = 1.f / (float)((size_t)BATCH * NPTS);

    prep_kernel<<<(BATCH * NPTS + 255) / 256, 256, 0, stream>>>(x0, w1, x0t, wp);

    struct Layer {
        const float* xin; int C, O;
        const float *W, *g, *bb; float* xout;
    } L[4] = {
        { x0t, 4,   64,  wp, g1, b1, x1 },
        { x1,  64,  64,  w2, g2, b2, x2 },
        { x2,  64,  128, w3, g3, b3, x3 },
        { x3,  128, 256, w4, g4, b4, x4 },
    };

    for (int li = 0; li < 4; ++li) {
        const int C = L[li].C, O = L[li].O;

        // 1. KNN on this layer's input features
        if (C == 4)       knn_kernel<4>  <<<BATCH * 8, 128, 0, stream>>>(L[li].xin, nb);
        else if (C == 64) knn_kernel<64> <<<BATCH * 8, 128, 0, stream>>>(L[li].xin, nb);
        else              knn_kernel<128><<<BATCH * 8, 128, 0, stream>>>(L[li].xin, nb);

        const int waves  = BATCH * (NKTOT / 32) * (O / 16);
        const int blocks = waves / 4;   // 4 waves / 128 threads per block

        // 2. BN statistics pass
        hipMemsetAsync(s0, 0, O * sizeof(float), stream);
        hipMemsetAsync(s1, 0, O * sizeof(float), stream);
        if (C == 4)
            edgeconv_gemm<4><<<blocks, 128, 0, stream>>>(L[li].xin, nb, L[li].W,
                nullptr, nullptr, s0, s1, nullptr, O, 0);
        else if (C == 64)
            edgeconv_gemm<64><<<blocks, 128, 0, stream>>>(L[li].xin, nb, L[li].W,
                nullptr, nullptr, s0, s1, nullptr, O, 0);
        else
            edgeconv_gemm<128><<<blocks, 128, 0, stream>>>(L[li].xin, nb, L[li].W,
                nullptr, nullptr, s0, s1, nullptr, O, 0);

        // 3. finalize scale/shift
        bn_finalize<<<(O + 255) / 256, 256, 0, stream>>>(s0, s1, L[li].g, L[li].bb,
                                                         sc, sh, O, inv_nk);

        // 4. apply: recompute, BN+ReLU, max over K
        hipMemsetAsync(L[li].xout, 0, (size_t)BATCH * NPTS * O * sizeof(float), stream);
        if (C == 4)
            edgeconv_gemm<4><<<blocks, 128, 0, stream>>>(L[li].xin, nb, L[li].W,
                sc, sh, nullptr, nullptr, (unsigned int*)L[li].xout, O, 1);
        else if (C == 64)
            edgeconv_gemm<64><<<blocks, 128, 0, stream>>>(L[li].xin, nb, L[li].W,
                sc, sh, nullptr, nullptr, (unsigned int*)L[li].xout, O, 1);
        else
            edgeconv_gemm<128><<<blocks, 128, 0, stream>>>(L[li].xin, nb, L[li].W,
                sc, sh, nullptr, nullptr, (unsigned int*)L[li].xout, O, 1);
    }

    // Final projection: 1024 x 512 over B*N columns
    const int waves5  = 64 * (BATCH * NPTS / 32);   // 16384
    const int blocks5 = waves5 / 4;

    hipMemsetAsync(s0, 0, 1024 * sizeof(float), stream);
    hipMemsetAsync(s1, 0, 1024 * sizeof(float), stream);
    final_gemm<<<blocks5, 128, 0, stream>>>(x1, x2, x3, x4, w5,
                                            nullptr, nullptr, s0, s1, nullptr, 0);
    bn_finalize<<<4, 256, 0, stream>>>(s0, s1, g5, b5, sc, sh, 1024, inv_n);
    hipMemsetAsync(d_out, 0, (size_t)BATCH * 1024 * sizeof(float), stream);
    final_gemm<<<blocks5, 128, 0, stream>>>(x1, x2, x3, x4, w5,
                                            sc, sh, nullptr, nullptr,
                                            (unsigned int*)d_out, 1);
}